// pool_model_45466523796147
// MI455X (gfx1250) — compile-verified
//
#include <hip/hip_runtime.h>
#include <hip/hip_bf16.h>

// CDNA5 / gfx1250: fused masked-mean + global-max pool over x[64,512,784].
// masked_sum via V_WMMA_F32_16X16X4_F32: A = mask tile (rows identical),
// B = 4x16 x-tile (N = channel across lanes 0..15). Spatial order inside each
// 8-wide chunk is permuted so every lane issues one contiguous b128 load:
//   lanes 0..15  load x[s..s+3]   -> K slots {0,1} of WMMA#0 / WMMA#1
//   lanes 16..31 load x[s+4..s+7] -> K slots {2,3} of WMMA#0 / WMMA#1
// The mask operand uses the identical permutation, so D accumulates
// sum_s x[c,s]*m[s] exactly. Two waves split the spatial range (2x occupancy),
// combining 16 sums + 16 maxes + count through LDS.

typedef __attribute__((ext_vector_type(2))) float v2f;
typedef __attribute__((ext_vector_type(4))) float v4f;
typedef __attribute__((ext_vector_type(8))) float v8f;

#define BDIM 64  // 2 waves per block, one per spatial half

__global__ __launch_bounds__(BDIM) void masked_pool_kernel(
    const float* __restrict__ x,          // [64, 512, 784] f32
    const unsigned char* __restrict__ cc, // [64, 784] bool bytes (0/1)
    float* __restrict__ out)              // [64, 1024] f32
{
    constexpr int C  = 512;
    constexpr int S  = 28 * 28;   // 784
    constexpr int SH = S / 2;     // 392 = 49 * 8

    const int wv   = threadIdx.x >> 5;     // spatial half (0/1)
    const int lane = threadIdx.x & 31;
    const int b    = blockIdx.x >> 5;      // 32 channel-groups per batch
    const int c0   = (blockIdx.x & 31) << 4;
    const int n    = lane & 15;            // N index = channel-in-group
    const int q    = (lane >> 4) & 1;      // 0 -> K{0,1}, 1 -> K{2,3}

    const float* __restrict__ xrow = x + (size_t)(b * C + c0 + n) * S;
    const unsigned char* __restrict__ mrow = cc + (size_t)b * S;

    const int sBeg = wv * SH;

    v8f acc = {};
    float vmax = -3.402823466e+38f;
    unsigned int mcnt = 0;

    #pragma unroll 7
    for (int s = sBeg; s < sBeg + SH; s += 8) {   // 49 iterations, 7x7 exact
        const int o = s + 4 * q;
        // 16B contiguous per lane, 16B aligned (row stride 3136B, o mult of 4)
        v4f xv = *(const v4f*)(xrow + o);
        // broadcast 4 mask bytes for this lane-half's K slots
        unsigned int mv = *(const unsigned int*)(mrow + o);
        float m0 = (float)( mv        & 0xFFu);
        float m1 = (float)((mv >> 8 ) & 0xFFu);
        float m2 = (float)((mv >> 16) & 0xFFu);
        float m3 = (float)( mv >> 24        );

        v2f a0; a0.x = m0;   a0.y = m1;
        v2f b0; b0.x = xv.x; b0.y = xv.y;
        acc = __builtin_amdgcn_wmma_f32_16x16x4_f32(
                  false, a0, false, b0, (short)0, acc, false, false);

        v2f a1; a1.x = m2;   a1.y = m3;
        v2f b1; b1.x = xv.z; b1.y = xv.w;
        acc = __builtin_amdgcn_wmma_f32_16x16x4_f32(
                  false, a1, false, b1, (short)0, acc, false, false);

        vmax = fmaxf(fmaxf(vmax, fmaxf(xv.x, xv.y)), fmaxf(xv.z, xv.w));
        mcnt += (mv & 0xFFu) + ((mv >> 8) & 0xFFu)
              + ((mv >> 16) & 0xFFu) + (mv >> 24);
    }

    // Each mask byte in this wave's range was read by 16 lanes -> sum/16.
    #pragma unroll
    for (int off = 16; off > 0; off >>= 1)
        mcnt += __shfl_xor(mcnt, off, 32);
    float cnt = (float)(mcnt >> 4);

    // lanes l and l^16 covered complementary K slots of the same channel
    vmax = fmaxf(vmax, __shfl_xor(vmax, 16, 32));

    // Cross-wave combine: 16 sums + 16 maxes + count per wave.
    __shared__ float lsum[2][16];
    __shared__ float lmax[2][16];
    __shared__ float lcnt[2];
    if (lane < 16) {
        lsum[wv][n] = acc[0];   // D row M=0 (all mask rows identical)
        lmax[wv][n] = vmax;
    }
    if (lane == 0) lcnt[wv] = cnt;
    __syncthreads();

    if (threadIdx.x < 16) {    // wave 0, lanes 0..15; xrow is channel c0+n row
        float total = lsum[0][n] + lsum[1][n] + xrow[0];  // + x[:,:,0,0]
        float cntT  = lcnt[0] + lcnt[1];
        float denom = (cntT == 0.0f) ? 1.0f : cntT;
        float* ob = out + (size_t)b * (2 * C);
        ob[c0 + n]     = total / denom;                   // mean_pool half
        ob[C + c0 + n] = fmaxf(lmax[0][n], lmax[1][n]);   // max_pool half
    }
}

extern "C" void kernel_launch(void* const* d_in, const int* in_sizes, int n_in,
                              void* d_out, int out_size, void* d_ws, size_t ws_size,
                              hipStream_t stream) {
    (void)in_sizes; (void)n_in; (void)out_size; (void)d_ws; (void)ws_size;
    const float* x = (const float*)d_in[0];
    const unsigned char* cc = (const unsigned char*)d_in[1];
    float* out = (float*)d_out;

    // 64 batches * 32 channel-groups = 2048 blocks of 2 waves (4096 waves).
    dim3 grid(64 * 32);
    dim3 block(BDIM);
    hipLaunchKernelGGL(masked_pool_kernel, grid, block, 0, stream, x, cc, out);
}